// LoraLayer_40819369181424
// MI455X (gfx1250) — compile-verified
//
#include <hip/hip_runtime.h>
#include <hip/hip_bf16.h>

// LoRA grouped-GEMM for MI455X (gfx1250), fp32 via V_WMMA_F32_16X16X4_F32.
// Phase 1: inter[m, tok, r] = (x[sorted_ids[tok]] @ A[m, slot(tok)])[r], rank-masked.
// Phase 2: out[tok, m*OUT + o] = sum_r inter[m, tok, r] * B[m, slot(tok), r, o].
//
// v2: 32-bit element offsets from SGPR bases (saddr + voffset addressing,
// constants folded into the 24-bit immediate field) and software-pipelined
// fragment loads so WMMAs overlap outstanding loads instead of waiting on
// loadcnt==0 every step.

typedef float v2f __attribute__((ext_vector_type(2)));
typedef float v8f __attribute__((ext_vector_type(8)));

namespace {
constexpr int BS         = 4096;
constexpr int HIDDEN     = 2048;
constexpr int MAX_RANK   = 16;
constexpr int N_SLOTS    = 8;
constexpr int N_MOD      = 2;
constexpr int OUT        = 8192;
constexpr int OUT_STRIDE = N_MOD * OUT;   // 16384 columns in d_out
}

__device__ __forceinline__ v8f wmma_f32_k4(v2f a, v2f b, v8f c) {
  return __builtin_amdgcn_wmma_f32_16x16x4_f32(false, a, false, b,
                                               (short)0, c, false, false);
}

// ---------------------------------------------------------------------------
// Phase 1: 8 waves / block, each wave owns one 16-token tile for one module.
// WMMA A-frag = x tile (16 tokens x 4 hidden), B-frag = LoRA-A chunk (4 x 16).
// ---------------------------------------------------------------------------
__global__ void lora_phase1_wmma(const float* __restrict__ x,
                                 const float* __restrict__ A,
                                 const int*   __restrict__ sorted_ids,
                                 const int*   __restrict__ slot_ranks,
                                 const int*   __restrict__ slot_offsets,
                                 float*       __restrict__ inter) {
  const int lane = threadIdx.x & 31;
  const int wave = threadIdx.x >> 5;                 // 0..7
  const int m    = blockIdx.y;
  const int tile = blockIdx.x * 8 + wave;            // 16-token tile, 0..255
  const int tok0 = tile * 16;

  // slot of this tile (slot boundaries are multiples of 512 -> uniform per tile)
  int slot = 0;
  for (int s = 1; s < N_SLOTS; ++s)
    if (slot_offsets[s] <= tok0) slot = s;

  const int half = lane >> 4;                        // 0 | 1
  const int l15  = lane & 15;

  // A-frag row for this lane: token M = l15 (gathered through sorted_ids).
  const int row = sorted_ids[tok0 + l15];

  // 32-bit element offsets from the SGPR kernel-arg bases.
  unsigned xoff = (unsigned)row * HIDDEN + 2u * half;
  unsigned aoff = ((unsigned)(m * N_SLOTS + slot) * HIDDEN + 2u * half) * MAX_RANK + l15;

  // Prologue: fragments for k = 0.
  v2f af = *(const v2f*)(x + xoff);
  v2f bf;
  bf.x = A[aoff];
  bf.y = A[aoff + MAX_RANK];

  v8f acc = {};
#pragma unroll 8
  for (int k = 0; k < HIDDEN - 4; k += 4) {
    // Issue next iteration's loads before consuming the current fragments.
    xoff += 4;
    aoff += 4 * MAX_RANK;
    v2f af_n = *(const v2f*)(x + xoff);
    v2f bf_n;
    bf_n.x = A[aoff];
    bf_n.y = A[aoff + MAX_RANK];

    acc = wmma_f32_k4(af, bf, acc);
    af = af_n;
    bf = bf_n;
  }
  acc = wmma_f32_k4(af, bf, acc);                    // epilogue step

  // D 16x16 f32 layout: VGPR j -> M = j + 8*half, N = l15 (N == rank index).
  const int   rank = slot_ranks[slot];
  const float rmsk = (l15 < rank) ? 1.0f : 0.0f;
  unsigned ioff = ((unsigned)m * BS + (unsigned)(tok0 + 8 * half)) * MAX_RANK + l15;
#pragma unroll
  for (int j = 0; j < 8; ++j)
    inter[ioff + (unsigned)j * MAX_RANK] = acc[j] * rmsk;
}

// ---------------------------------------------------------------------------
// Phase 2: block = 16 tokens x 8192 cols for one module; 8 waves, each sweeps
// a 16-col phase. A-frags (inter) loaded once per wave; B-frags double-
// buffered so WMMA + NT stores overlap the next tile's loads.
// ---------------------------------------------------------------------------
__global__ void lora_phase2_wmma(const float* __restrict__ B,
                                 const float* __restrict__ inter,
                                 const int*   __restrict__ slot_offsets,
                                 float*       __restrict__ out) {
  const int lane = threadIdx.x & 31;
  const int wave = threadIdx.x >> 5;                 // 0..7 -> column phase
  const int m    = blockIdx.y;
  const int tok0 = blockIdx.x * 16;
  const int half = lane >> 4;
  const int l15  = lane & 15;

  int slot = 0;
  for (int s = 1; s < N_SLOTS; ++s)
    if (slot_offsets[s] <= tok0) slot = s;

  // A-frags over the full rank-16 K dimension: 4 WMMA K-steps, reused for all
  // 64 column tiles handled by this wave.
  const unsigned iof =
      ((unsigned)m * BS + (unsigned)(tok0 + l15)) * MAX_RANK + 2u * half;
  v2f af[4];
#pragma unroll
  for (int kk = 0; kk < 4; ++kk)
    af[kk] = *(const v2f*)(inter + iof + kk * 4);

  const int cb0 = wave * 16;
  // B element offset: constants kk*4*OUT and +OUT fold into immediate offsets.
  unsigned bof =
      ((unsigned)(m * N_SLOTS + slot) * MAX_RANK + 2u * half) * OUT + cb0 + l15;
  // Output element offset: j*OUT_STRIDE folds into immediate offsets.
  unsigned oof =
      (unsigned)(tok0 + 8 * half) * OUT_STRIDE + (unsigned)m * OUT + cb0 + l15;

  // Prologue: B-frags for the first column tile.
  v2f bf[4];
#pragma unroll
  for (int kk = 0; kk < 4; ++kk) {
    bf[kk].x = B[bof + kk * 4 * OUT];
    bf[kk].y = B[bof + kk * 4 * OUT + OUT];
  }

  constexpr int NSTEP = OUT / (8 * 16);              // 64 column tiles per wave
  for (int it = 0; it < NSTEP - 1; ++it) {
    // Next tile's B-frags in flight while this tile computes and stores.
    const unsigned bn = bof + 8 * 16;
    v2f bfn[4];
#pragma unroll
    for (int kk = 0; kk < 4; ++kk) {
      bfn[kk].x = B[bn + kk * 4 * OUT];
      bfn[kk].y = B[bn + kk * 4 * OUT + OUT];
    }

    v8f acc = {};
#pragma unroll
    for (int kk = 0; kk < 4; ++kk)
      acc = wmma_f32_k4(af[kk], bf[kk], acc);

#pragma unroll
    for (int j = 0; j < 8; ++j)
      __builtin_nontemporal_store(acc[j], out + oof + (unsigned)j * OUT_STRIDE);

#pragma unroll
    for (int kk = 0; kk < 4; ++kk) bf[kk] = bfn[kk];
    bof = bn;
    oof += 8 * 16;
  }

  // Epilogue tile.
  v8f acc = {};
#pragma unroll
  for (int kk = 0; kk < 4; ++kk)
    acc = wmma_f32_k4(af[kk], bf[kk], acc);
#pragma unroll
  for (int j = 0; j < 8; ++j)
    __builtin_nontemporal_store(acc[j], out + oof + (unsigned)j * OUT_STRIDE);
}

// ---------------------------------------------------------------------------
extern "C" void kernel_launch(void* const* d_in, const int* in_sizes, int n_in,
                              void* d_out, int out_size, void* d_ws, size_t ws_size,
                              hipStream_t stream) {
  (void)in_sizes; (void)n_in; (void)out_size; (void)ws_size;
  const float* x            = (const float*)d_in[0];
  const float* A            = (const float*)d_in[1];
  const float* B            = (const float*)d_in[2];
  const int*   sorted_ids   = (const int*)d_in[3];
  // d_in[4] = slot_counts (offsets already provided)
  const int*   slot_ranks   = (const int*)d_in[5];
  const int*   slot_offsets = (const int*)d_in[6];
  float*       out          = (float*)d_out;
  float*       inter        = (float*)d_ws;          // [N_MOD, BS, MAX_RANK] = 512 KB

  dim3 g1(BS / 16 / 8, N_MOD);                       // (32, 2), 8 waves/block
  lora_phase1_wmma<<<g1, 256, 0, stream>>>(x, A, sorted_ids, slot_ranks,
                                           slot_offsets, inter);

  dim3 g2(BS / 16, N_MOD);                           // (256, 2), 8 waves/block
  lora_phase2_wmma<<<g2, 256, 0, stream>>>(B, inter, slot_offsets, out);
}